// STN_65395172049489
// MI455X (gfx1250) — compile-verified
//
#include <hip/hip_runtime.h>
#include <hip/hip_bf16.h>

// ---------------------------------------------------------------------------
// STN inference for MI455X (gfx1250, wave32).
//   Stage 1: C1[64,64] = X[64,K] @ W1[K,64]  (K = 196608) via V_WMMA_F32_16X16X4_F32
//            192 K-chunk blocks -> per-block partial tiles -> fixed-order reduce
//            (deterministic; no float atomics on the primary path)
//   Stage 2: mat[64,6] = theta * relu(relu(C1+b1) @ W2 + b2)
//   Stage 3: bilinear sampling of the affine grid
// ---------------------------------------------------------------------------

typedef __attribute__((ext_vector_type(2))) float v2f;
typedef __attribute__((ext_vector_type(8))) float v8f;

#define K_TOTAL  196608   // 256*256*3
#define K_BLOCKS 192
#define K_CHUNK  1024     // 196608 / 192

// ---------------------------------------------------------------------------
// zero the 64x64 accumulator (atomic-fallback path only)
// ---------------------------------------------------------------------------
__global__ __launch_bounds__(256) void stn_zero_acc(float* __restrict__ acc) {
    int i = blockIdx.x * 256 + threadIdx.x;
    if (i < 64 * 64) acc[i] = 0.0f;
}

// ---------------------------------------------------------------------------
// GEMM1: each block owns one K-chunk; 16 waves cover the 4x4 grid of 16x16
// output tiles. ATOMIC=false: store partial tile to part[block]. ATOMIC=true:
// fallback global_atomic_add_f32 into a single accumulator.
//
// fp32 WMMA fragment layout (ISA 7.12.2):
//   A (16x4):  lanes 0-15 -> M=lane, VGPR0=K0, VGPR1=K1 ; lanes 16-31 -> K2,K3
//   B (4x16):  lanes 0-15 -> N=lane, VGPR0=K0, VGPR1=K1 ; lanes 16-31 -> K2,K3
//   C (16x16): VGPR v, lanes 0-15 -> M=v, N=lane ; lanes 16-31 -> M=v+8
// ---------------------------------------------------------------------------
template <bool ATOMIC>
__global__ __launch_bounds__(512) void stn_gemm1(const float* __restrict__ X,
                                                 const float* __restrict__ W1,
                                                 float* __restrict__ outp) {
    const int wave = threadIdx.x >> 5;        // 0..15
    const int lane = threadIdx.x & 31;
    const int mt = (wave >> 2) * 16;          // M tile base (batch rows)
    const int nt = (wave & 3) * 16;           // N tile base (hidden units)
    const int l15 = lane & 15;
    const int kg  = (lane >> 4) * 2;          // 0 or 2: K pair handled by this half-wave

    const long k0 = (long)blockIdx.x * K_CHUNK;

    const float* Arow = X  + (long)(mt + l15) * K_TOTAL + k0 + kg;  // A: row m, 2 consecutive K
    const float* Bcol = W1 + (k0 + kg) * 64 + (nt + l15);           // B: col n, stride-64 K

    v8f c0 = {}, c1 = {}, c2 = {}, c3 = {};

    for (int kk = 0; kk < K_CHUNK; kk += 16) {
        // ---- K offset kk+0 ----
        v2f a0 = *(const v2f*)(Arow + kk);
        v2f b0; b0.x = Bcol[(long)(kk + 0) * 64]; b0.y = Bcol[(long)(kk + 1) * 64];
        c0 = __builtin_amdgcn_wmma_f32_16x16x4_f32(false, a0, false, b0, (short)0, c0, false, false);
        // ---- K offset kk+4 ----
        v2f a1 = *(const v2f*)(Arow + kk + 4);
        v2f b1v; b1v.x = Bcol[(long)(kk + 4) * 64]; b1v.y = Bcol[(long)(kk + 5) * 64];
        c1 = __builtin_amdgcn_wmma_f32_16x16x4_f32(false, a1, false, b1v, (short)0, c1, false, false);
        // ---- K offset kk+8 ----
        v2f a2 = *(const v2f*)(Arow + kk + 8);
        v2f b2v; b2v.x = Bcol[(long)(kk + 8) * 64]; b2v.y = Bcol[(long)(kk + 9) * 64];
        c2 = __builtin_amdgcn_wmma_f32_16x16x4_f32(false, a2, false, b2v, (short)0, c2, false, false);
        // ---- K offset kk+12 ----
        v2f a3 = *(const v2f*)(Arow + kk + 12);
        v2f b3; b3.x = Bcol[(long)(kk + 12) * 64]; b3.y = Bcol[(long)(kk + 13) * 64];
        c3 = __builtin_amdgcn_wmma_f32_16x16x4_f32(false, a3, false, b3, (short)0, c3, false, false);
    }

    // combine the 4 accumulator chains; emit this block's 64x64 partial tile
    const int crow = mt + ((lane >> 4) ? 8 : 0);
    const int ccol = nt + l15;
#pragma unroll
    for (int v = 0; v < 8; ++v) {
        float s = c0[v] + c1[v] + c2[v] + c3[v];
        int idx = (crow + v) * 64 + ccol;
        if (ATOMIC) {
            atomicAdd(&outp[idx], s);
        } else {
            outp[(long)blockIdx.x * 4096 + idx] = s;
        }
    }
}

// ---------------------------------------------------------------------------
// Fixed-order reduction of the 192 partial tiles -> acc[64*64] (deterministic)
// ---------------------------------------------------------------------------
__global__ __launch_bounds__(256) void stn_reduce(const float* __restrict__ part,
                                                  float* __restrict__ acc) {
    int i = blockIdx.x * 256 + threadIdx.x;   // < 4096
    float s = 0.0f;
#pragma unroll 8
    for (int j = 0; j < K_BLOCKS; ++j) s += part[(long)j * 4096 + i];
    acc[i] = s;
}

// ---------------------------------------------------------------------------
// Head: bias+relu, (64,64)@(64,6)+bias+relu, scale by theta -> mat[64,6]
// ---------------------------------------------------------------------------
__global__ __launch_bounds__(64) void stn_head(const float* __restrict__ acc,
                                               const float* __restrict__ b1,
                                               const float* __restrict__ W2,
                                               const float* __restrict__ b2,
                                               const float* __restrict__ theta,
                                               float* __restrict__ mat) {
    int b = threadIdx.x;
    if (b >= 64) return;
    float h[64];
#pragma unroll 8
    for (int i = 0; i < 64; ++i) h[i] = fmaxf(acc[b * 64 + i] + b1[i], 0.0f);
#pragma unroll
    for (int j = 0; j < 6; ++j) {
        float s = b2[j];
#pragma unroll 8
        for (int i = 0; i < 64; ++i) s += h[i] * W2[i * 6 + j];
        mat[b * 6 + j] = theta[b * 6 + j] * fmaxf(s, 0.0f);
    }
}

// ---------------------------------------------------------------------------
// Bilinear sampler over the affine grid. One thread per output pixel (3 ch).
// ---------------------------------------------------------------------------
__global__ __launch_bounds__(256) void stn_sample(const float* __restrict__ img,
                                                  const float* __restrict__ mat,
                                                  float* __restrict__ out) {
    int p = blockIdx.x * 256 + threadIdx.x;       // 64*256*256 pixels
    int b = p >> 16;
    int y = (p >> 8) & 255;
    int x = p & 255;

    const float* m = mat + b * 6;
    float xg = x * (2.0f / 255.0f) - 1.0f;
    float yg = y * (2.0f / 255.0f) - 1.0f;
    float xs = m[0] * xg + m[1] * yg + m[2];
    float ys = m[3] * xg + m[4] * yg + m[5];

    float xf = 0.5f * (xs + 1.0f) * 255.0f;
    float yf = 0.5f * (ys + 1.0f) * 255.0f;
    int x0 = min(max((int)floorf(xf), 0), 255);
    int y0 = min(max((int)floorf(yf), 0), 255);
    int x1 = min(x0 + 1, 255);
    int y1 = min(y0 + 1, 255);

    float x0f = (float)x0, x1f = (float)x1, y0f = (float)y0, y1f = (float)y1;
    float wa = (x1f - xf) * (y1f - yf);
    float wb = (x1f - xf) * (yf - y0f);
    float wc = (xf - x0f) * (y1f - yf);
    float wd = (xf - x0f) * (yf - y0f);

    const float* base = img + ((long)b << 16) * 3;
    const float* Ia = base + (y0 * 256 + x0) * 3;
    const float* Ib = base + (y1 * 256 + x0) * 3;
    const float* Ic = base + (y0 * 256 + x1) * 3;
    const float* Id = base + (y1 * 256 + x1) * 3;

    float* o = out + (long)p * 3;
#pragma unroll
    for (int c = 0; c < 3; ++c)
        o[c] = wa * Ia[c] + wb * Ib[c] + wc * Ic[c] + wd * Id[c];
}

// ---------------------------------------------------------------------------
extern "C" void kernel_launch(void* const* d_in, const int* in_sizes, int n_in,
                              void* d_out, int out_size, void* d_ws, size_t ws_size,
                              hipStream_t stream) {
    const float* X     = (const float*)d_in[0];   // (64,256,256,3)
    const float* W1    = (const float*)d_in[1];   // (196608,64)
    const float* b1    = (const float*)d_in[2];   // (64,)
    const float* W2    = (const float*)d_in[3];   // (64,6)
    const float* b2    = (const float*)d_in[4];   // (6,)
    const float* theta = (const float*)d_in[5];   // (64,6)
    float* out = (float*)d_out;

    const size_t part_floats = (size_t)K_BLOCKS * 4096;
    const size_t need_full   = (part_floats + 4096 + 384) * sizeof(float);

    if (ws_size >= need_full) {
        // Primary path: deterministic partials + fixed-order reduction.
        float* part = (float*)d_ws;               // 192 x 64x64 partial tiles
        float* acc  = part + part_floats;         // 64x64 reduced C1
        float* mat  = acc + 4096;                 // 64x6 affine matrices

        stn_gemm1<false><<<K_BLOCKS, 512, 0, stream>>>(X, W1, part);
        stn_reduce<<<16, 256, 0, stream>>>(part, acc);
        stn_head<<<1, 64, 0, stream>>>(acc, b1, W2, b2, theta, mat);
        stn_sample<<<(64 * 256 * 256) / 256, 256, 0, stream>>>(X, mat, out);
    } else {
        // Fallback for tiny workspace: atomic accumulation.
        float* acc = (float*)d_ws;                // 64x64 accumulator
        float* mat = acc + 4096;                  // 64x6 affine matrices

        stn_zero_acc<<<16, 256, 0, stream>>>(acc);
        stn_gemm1<true><<<K_BLOCKS, 512, 0, stream>>>(X, W1, acc);
        stn_head<<<1, 64, 0, stream>>>(acc, b1, W2, b2, theta, mat);
        stn_sample<<<(64 * 256 * 256) / 256, 256, 0, stream>>>(X, mat, out);
    }
}